// EdgeGNN_27779848470880
// MI455X (gfx1250) — compile-verified
//
#include <hip/hip_runtime.h>

typedef __attribute__((ext_vector_type(2))) float v2f;
typedef __attribute__((ext_vector_type(8))) float v8f;

#define N_NODES 50000
#define N_EDGES 640000
#define DDIM    128
#define LN_EPS  1e-5f

// sB: B-fragment-packed weights. Indexed [kk][half][col] in float2 units.
// Half-stride 144 float2 = 288 dwords = 32 (mod 64) banks, so the two
// half-waves' contiguous b64 reads land on disjoint bank halves.
#define SB_HSTRIDE 144            // float2 per (kk,half) row
#define SB_SIZE    (32 * 2 * SB_HSTRIDE)   // float2 count

// sZ: activation tile, stride 132 floats: +1 row -> +4 banks. Row-wise LN
// reads (16 rows, same col) hit 16 distinct banks; A-fragment b64 reads of
// the two half-waves interleave conflict-free.
#define SZ_STRIDE 132

// ---------------------------------------------------------------------------
// agg = 0
// ---------------------------------------------------------------------------
__global__ void zero_f32(float* __restrict__ p, int n) {
    int i = blockIdx.x * blockDim.x + threadIdx.x;
    if (i < n) p[i] = 0.0f;
}

// ---------------------------------------------------------------------------
// agg[dst] += h[src]   (one wave per edge, float4 per lane, f32 atomics;
// agg is 25.6MB -> resident in 192MB L2, atomics resolve there)
// ---------------------------------------------------------------------------
__global__ __launch_bounds__(256) void scatter_add(
    const float* __restrict__ h, const int* __restrict__ ei,
    float* __restrict__ agg) {
    int gid  = blockIdx.x * blockDim.x + threadIdx.x;
    int e    = gid >> 5;
    int lane = gid & 31;
    if (e >= N_EDGES) return;
    int src = ei[e];
    int dst = ei[N_EDGES + e];
    const float4 v = *(const float4*)(h + (size_t)src * DDIM + lane * 4);
    float* a = agg + (size_t)dst * DDIM + lane * 4;
    atomicAdd(a + 0, v.x);
    atomicAdd(a + 1, v.y);
    atomicAdd(a + 2, v.z);
    atomicAdd(a + 3, v.w);
}

// ---------------------------------------------------------------------------
// Fused GIN MLP layer: z=(1+eps)h+agg ; z@W1+b1 ; ReLU ; @W2+b2 ; LN ;
// [skipLN ; ReLU] ; +h_in.  4 waves/block, 16 node-rows per wave.
// fp32 WMMA 16x16x4 per ISA 7.12.2 layouts; B staged fragment-packed.
// ---------------------------------------------------------------------------
__global__ __launch_bounds__(128) void gin_mlp(
    const float* __restrict__ h_in, const float* __restrict__ agg,
    const float* __restrict__ eps_arr, int layer, int has_skip,
    const float* __restrict__ W1, const float* __restrict__ b1,
    const float* __restrict__ W2, const float* __restrict__ b2,
    const float* __restrict__ lng, const float* __restrict__ lnb,
    const float* __restrict__ sg,  const float* __restrict__ sb,
    float* __restrict__ h_out) {

    __shared__ float2 sB[SB_SIZE];              // 73,728 B
    __shared__ float  sZ[4 * 16 * SZ_STRIDE];   // 33,792 B
    __shared__ float  sStat[4 * 16 * 2];        //    512 B (per-row mu2,inv2)

    const int tid   = threadIdx.x;
    const int wave  = tid >> 5;
    const int lane  = tid & 31;
    const int lhalf = lane >> 4;   // 0: K lo pair / rows 0..7 ; 1: K hi pair / rows 8..15
    const int l15   = lane & 15;

    const float* W1l = W1 + (size_t)layer * DDIM * DDIM;
    const float* W2l = W2 + (size_t)layer * DDIM * DDIM;
    const float* b1l = b1 + (size_t)layer * DDIM;
    const float* b2l = b2 + (size_t)layer * DDIM;
    const float* lgl = lng + (size_t)layer * DDIM;
    const float* lbl = lnb + (size_t)layer * DDIM;
    const float* sgl = sg + (size_t)layer * DDIM;   // only read when has_skip
    const float* sbl = sb + (size_t)layer * DDIM;

    const float epsv = 1.0f + eps_arr[layer];

    const int node_base = blockIdx.x * 64 + wave * 16;

    // A-fragment source row (clamped so WMMA runs with full EXEC on the tail)
    int arow = node_base + l15;
    if (arow >= N_NODES) arow = N_NODES - 1;
    const float* hrow = h_in + (size_t)arow * DDIM;
    const float* grow = agg  + (size_t)arow * DDIM;

    // ---- stage W1 fragment-packed: sB[kk][h][col] = {W[4kk+2h][col], W[4kk+2h+1][col]}
    for (int idx = tid; idx < 32 * 2 * DDIM; idx += 128) {
        const int kk   = idx >> 8;
        const int lh   = (idx >> 7) & 1;
        const int col  = idx & 127;
        const int krow = kk * 4 + lh * 2;
        float2 f;
        f.x = W1l[krow * DDIM + col];
        f.y = W1l[(krow + 1) * DDIM + col];
        sB[(kk * 2 + lh) * SB_HSTRIDE + col] = f;
    }
    __syncthreads();

    const v8f vzero = {};
    v8f acc[8];
#pragma unroll
    for (int t = 0; t < 8; ++t) acc[t] = vzero;

    // ---- GEMM1: acc = ((1+eps)h + agg) @ W1 ----
    const float2* bfrag = &sB[lhalf * SB_HSTRIDE + l15];
    for (int kk = 0; kk < 32; ++kk) {
        const int c0 = kk * 4 + lhalf * 2;
        float2 hv = *(const float2*)(hrow + c0);
        float2 gv = *(const float2*)(grow + c0);
        v2f a;
        a.x = epsv * hv.x + gv.x;
        a.y = epsv * hv.y + gv.y;
        const float2* bk = bfrag + kk * (2 * SB_HSTRIDE);
#pragma unroll
        for (int nt = 0; nt < 8; ++nt) {
            float2 bf = bk[nt * 16];
            v2f b; b.x = bf.x; b.y = bf.y;
            acc[nt] = __builtin_amdgcn_wmma_f32_16x16x4_f32(
                false, a, false, b, (short)0, acc[nt], false, false);
        }
    }

    // ---- bias + ReLU, park h1 tile in LDS ----
    {
        float* zt = &sZ[wave * 16 * SZ_STRIDE];
#pragma unroll
        for (int nt = 0; nt < 8; ++nt) {
            const int col  = nt * 16 + l15;
            const float bi = b1l[col];
#pragma unroll
            for (int v = 0; v < 8; ++v) {
                float val = acc[nt][v] + bi;
                val = fmaxf(val, 0.0f);
                zt[(v + lhalf * 8) * SZ_STRIDE + col] = val;
            }
        }
    }
    __syncthreads();

    // ---- swap W2 into LDS (same packed layout) ----
    for (int idx = tid; idx < 32 * 2 * DDIM; idx += 128) {
        const int kk   = idx >> 8;
        const int lh   = (idx >> 7) & 1;
        const int col  = idx & 127;
        const int krow = kk * 4 + lh * 2;
        float2 f;
        f.x = W2l[krow * DDIM + col];
        f.y = W2l[(krow + 1) * DDIM + col];
        sB[(kk * 2 + lh) * SB_HSTRIDE + col] = f;
    }
    __syncthreads();

#pragma unroll
    for (int t = 0; t < 8; ++t) acc[t] = vzero;

    // ---- GEMM2: acc = h1 @ W2 (A from LDS, single b64 per fragment) ----
    {
        const float* zrow = &sZ[(wave * 16 + l15) * SZ_STRIDE];
        for (int kk = 0; kk < 32; ++kk) {
            const int c0 = kk * 4 + lhalf * 2;
            float2 av = *(const float2*)(zrow + c0);
            v2f a; a.x = av.x; a.y = av.y;
            const float2* bk = bfrag + kk * (2 * SB_HSTRIDE);
#pragma unroll
            for (int nt = 0; nt < 8; ++nt) {
                float2 bf = bk[nt * 16];
                v2f b; b.x = bf.x; b.y = bf.y;
                acc[nt] = __builtin_amdgcn_wmma_f32_16x16x4_f32(
                    false, a, false, b, (short)0, acc[nt], false, false);
            }
        }
    }

    // ---- bias, park y tile in LDS for row-wise LayerNorm ----
    {
        float* zt = &sZ[wave * 16 * SZ_STRIDE];
#pragma unroll
        for (int nt = 0; nt < 8; ++nt) {
            const int col  = nt * 16 + l15;
            const float bi = b2l[col];
#pragma unroll
            for (int v = 0; v < 8; ++v)
                zt[(v + lhalf * 8) * SZ_STRIDE + col] = acc[nt][v] + bi;
        }
    }
    __syncthreads();

    // ---- Phase 1: LN stats + z1 in LDS; lanes (l, l+16) split each row ----
    {
        float* yrow = &sZ[(wave * 16 + l15) * SZ_STRIDE];
        const int cbase = lhalf * 64;

        float s = 0.0f, s2 = 0.0f;
#pragma unroll 4
        for (int j = 0; j < 64; ++j) {
            float y = yrow[cbase + j];
            s += y; s2 += y * y;
        }
        s  += __shfl_xor(s, 16, 32);
        s2 += __shfl_xor(s2, 16, 32);
        const float mu  = s * (1.0f / DDIM);
        const float var = s2 * (1.0f / DDIM) - mu * mu;
        const float inv = rsqrtf(var + LN_EPS);

        float t = 0.0f, t2 = 0.0f;
#pragma unroll 4
        for (int j = 0; j < 64; ++j) {
            const int c = cbase + j;
            float z1 = (yrow[c] - mu) * inv * lgl[c] + lbl[c];
            yrow[c] = z1;
            t += z1; t2 += z1 * z1;
        }

        if (has_skip) {
            t  += __shfl_xor(t, 16, 32);
            t2 += __shfl_xor(t2, 16, 32);
            float mu2  = t * (1.0f / DDIM);
            float var2 = t2 * (1.0f / DDIM) - mu2 * mu2;
            float inv2 = rsqrtf(var2 + LN_EPS);
            if (lhalf == 0) {
                sStat[(wave * 16 + l15) * 2 + 0] = mu2;
                sStat[(wave * 16 + l15) * 2 + 1] = inv2;
            }
        }
    }
    __syncthreads();

    // ---- Phase 2: coalesced skip-norm/ReLU + residual + store ----
    {
        const float* zt = &sZ[wave * 16 * SZ_STRIDE];
        const int c4 = lane * 4;                 // this lane's 4 columns
        float4 g4 = {1.f, 1.f, 1.f, 1.f}, sb4 = {0.f, 0.f, 0.f, 0.f};
        if (has_skip) {
            g4  = *(const float4*)(sgl + c4);
            sb4 = *(const float4*)(sbl + c4);
        }
#pragma unroll 4
        for (int r = 0; r < 16; ++r) {
            const int node = node_base + r;
            if (node >= N_NODES) break;
            const float* zr = zt + r * SZ_STRIDE + c4;
            float4 z = {zr[0], zr[1], zr[2], zr[3]};
            if (has_skip) {
                const float mu2  = sStat[(wave * 16 + r) * 2 + 0];
                const float inv2 = sStat[(wave * 16 + r) * 2 + 1];
                z.x = fmaxf((z.x - mu2) * inv2 * g4.x + sb4.x, 0.0f);
                z.y = fmaxf((z.y - mu2) * inv2 * g4.y + sb4.y, 0.0f);
                z.z = fmaxf((z.z - mu2) * inv2 * g4.z + sb4.z, 0.0f);
                z.w = fmaxf((z.w - mu2) * inv2 * g4.w + sb4.w, 0.0f);
            }
            const float4 hres = *(const float4*)(h_in + (size_t)node * DDIM + c4);
            float4 o;
            o.x = z.x + hres.x; o.y = z.y + hres.y;
            o.z = z.z + hres.z; o.w = z.w + hres.w;
            *(float4*)(h_out + (size_t)node * DDIM + c4) = o;
        }
    }
}

// ---------------------------------------------------------------------------
// Host-side launcher
// ---------------------------------------------------------------------------
extern "C" void kernel_launch(void* const* d_in, const int* in_sizes, int n_in,
                              void* d_out, int out_size, void* d_ws, size_t ws_size,
                              hipStream_t stream) {
    (void)in_sizes; (void)n_in; (void)out_size; (void)ws_size;

    const float* x   = (const float*)d_in[0];
    const int*   ei  = (const int*)d_in[1];
    const float* eps = (const float*)d_in[2];
    const float* W1  = (const float*)d_in[3];
    const float* b1  = (const float*)d_in[4];
    const float* W2  = (const float*)d_in[5];
    const float* b2  = (const float*)d_in[6];
    const float* lng = (const float*)d_in[7];
    const float* lnb = (const float*)d_in[8];
    const float* sg  = (const float*)d_in[9];
    const float* sb  = (const float*)d_in[10];
    float* out = (float*)d_out;

    const size_t feat = (size_t)N_NODES * DDIM;
    float* agg = (float*)d_ws;          // 25.6 MB
    float* hA  = agg + feat;            // 25.6 MB
    float* hB  = hA + feat;             // 25.6 MB

    const int zero_blocks    = (N_NODES * DDIM + 255) / 256;
    const int scatter_blocks = (N_EDGES * 32) / 256;
    const int mlp_blocks     = (N_NODES + 63) / 64;

    const float* hin = x;
    for (int i = 0; i < 3; ++i) {
        float* hout = (i == 0) ? hA : (i == 1) ? hB : out;
        const int has_skip = (i < 2) ? 1 : 0;

        zero_f32<<<zero_blocks, 256, 0, stream>>>(agg, N_NODES * DDIM);
        scatter_add<<<scatter_blocks, 256, 0, stream>>>(hin, ei, agg);
        gin_mlp<<<mlp_blocks, 128, 0, stream>>>(
            hin, agg, eps, i, has_skip,
            W1, b1, W2, b2, lng, lnb, sg, sb, hout);

        hin = hout;
    }
}